// GNNLayer_41334765256943
// MI455X (gfx1250) — compile-verified
//
#include <hip/hip_runtime.h>
#include <hip/hip_bf16.h>
#include <math.h>

#define HID   256
#define HEADS 8
#define DH    32

typedef float v2f __attribute__((ext_vector_type(2)));
typedef float v8f __attribute__((ext_vector_type(8)));

// Float atomic max via sign trick: non-negative floats order like ints,
// negative floats order reversed as unsigned. One atomic, no CAS loop.
__device__ __forceinline__ void atomicMaxF(float* addr, float val) {
    if (val >= 0.0f)
        atomicMax((int*)addr, __float_as_int(val));
    else
        atomicMin((unsigned int*)addr, __float_as_uint(val));
}

// ---------------------------------------------------------------------------
// 0) init: zero agg (=d_out) and denom, set segmax = -inf
// ---------------------------------------------------------------------------
__global__ void init_kernel(float* __restrict__ agg, float* __restrict__ segmax,
                            float* __restrict__ denom, int nAgg, int nSeg) {
    int t = blockIdx.x * blockDim.x + threadIdx.x;
    if (t < nAgg) agg[t] = 0.0f;
    if (t < nSeg) { segmax[t] = -INFINITY; denom[t] = 0.0f; }
}

// ---------------------------------------------------------------------------
// 1) Fused Q/K/V/skip GEMM via V_WMMA_F32_16X16X4_F32 (fp32 in, fp32 acc).
//    Block = 8 waves sharing ONE 16-row x tile staged in LDS via the CDNA5
//    async global->LDS path (ASYNCcnt). Each wave computes one 16x16 output
//    tile: ntile = blockIdx.y*8 + wave in [0,64); region = ntile>>4 selects
//    which of Wq/Wk/Wv/Wskip (B stays global: 1 MB total, L2-resident).
//    Requires N % 16 == 0 (true here: 50000 = 3125*16).
// ---------------------------------------------------------------------------
__global__ void qkvs_gemm_kernel(const float* __restrict__ x,
                                 const float* __restrict__ Wq, const float* __restrict__ bq,
                                 const float* __restrict__ Wk, const float* __restrict__ bk,
                                 const float* __restrict__ Wv, const float* __restrict__ bv,
                                 const float* __restrict__ Ws, const float* __restrict__ bs,
                                 float* __restrict__ q, float* __restrict__ k,
                                 float* __restrict__ v, float* __restrict__ sk,
                                 int Nn)
{
    __shared__ __align__(16) float atile[16 * HID];   // 16 KB x-tile

    const int wave = threadIdx.x >> 5;
    const int lane = threadIdx.x & 31;
    const int m0   = blockIdx.x << 4;

    // ---- async-stage x[m0..m0+15, :] into LDS (4096 floats = 1024 x b128) ----
    {
        const float* gsrc = x + (size_t)m0 * HID;
        #pragma unroll
        for (int it = 0; it < 4; ++it) {
            const int fidx = (it * 256 + threadIdx.x) * 4;            // float index
            const unsigned lds_off =
                (unsigned)(size_t)(const void*)&atile[fidx];          // LDS byte offset
            const unsigned long long gaddr =
                (unsigned long long)(size_t)(const void*)(gsrc + fidx);
            asm volatile("global_load_async_to_lds_b128 %0, %1, off"
                         :: "v"(lds_off), "v"(gaddr)
                         : "memory");
        }
        asm volatile("s_wait_asynccnt 0x0" ::: "memory");
    }
    __syncthreads();

    const int ntile  = (blockIdx.y << 3) + wave;  // 0..63
    const int region = ntile >> 4;                // 0..3 -> q,k,v,skip
    const int n0     = (ntile & 15) << 4;         // col base within [0,256)

    const float* W  = (region == 0) ? Wq : (region == 1) ? Wk : (region == 2) ? Wv : Ws;
    const float* bb = (region == 0) ? bq : (region == 1) ? bk : (region == 2) ? bv : bs;
    float*     outp = (region == 0) ? q  : (region == 1) ? k  : (region == 2) ? v  : sk;

    const int hl = lane >> 4;        // 0 -> K pair {0,1}; 1 -> K pair {2,3}
    const int ml = lane & 15;

    const float* arow = &atile[ml * HID + (hl << 1)];   // LDS (ds_load_b64)
    const float* bptr = W + n0 + ml;

    v8f acc = {};
    #pragma unroll 4
    for (int kk = 0; kk < HID; kk += 4) {
        // A 16x4 f32 fragment: VGPR0 = K=(kk+2*hl), VGPR1 = K+1, row M = m0+ml
        v2f a = *(const v2f*)(arow + kk);
        // B 4x16 f32 fragment: VGPR0 = row K=(kk+2*hl), VGPR1 = row K+1, col N = n0+ml
        const int krow = kk + (hl << 1);
        v2f b;
        b.x = bptr[(size_t)krow * HID];
        b.y = bptr[(size_t)(krow + 1) * HID];
        acc = __builtin_amdgcn_wmma_f32_16x16x4_f32(false, a, false, b,
                                                    (short)0, acc, false, false);
    }

    // C/D layout: VGPR r -> row m0 + r + 8*hl, col n0 + ml
    const int col   = n0 + ml;
    const float bia = bb[col];
    const int rbase = m0 + (hl << 3);
    #pragma unroll
    for (int r = 0; r < 8; ++r) {
        const int row = rbase + r;
        if (row < Nn) outp[(size_t)row * HID + col] = acc[r] + bia;
    }
}

// ---------------------------------------------------------------------------
// 2) Per-(edge,head) attention logit + segment max (L2-resident gathers)
// ---------------------------------------------------------------------------
__global__ void edge_logits_kernel(const int* __restrict__ ei,
                                   const float* __restrict__ q, const float* __restrict__ k,
                                   float* __restrict__ logits, float* __restrict__ segmax,
                                   int Ee)
{
    int t = blockIdx.x * blockDim.x + threadIdx.x;
    if (t >= Ee * HEADS) return;
    const int e = t >> 3;
    const int h = t & 7;
    const int src = ei[e];
    const int dst = ei[Ee + e];
    const float4* qr = (const float4*)(q + (size_t)dst * HID + h * DH);
    const float4* kr = (const float4*)(k + (size_t)src * HID + h * DH);
    float s = 0.0f;
    #pragma unroll
    for (int i = 0; i < DH / 4; ++i) {
        float4 a = qr[i], b = kr[i];
        s += a.x * b.x + a.y * b.y + a.z * b.z + a.w * b.w;
    }
    s *= 0.17677669529663687f;   // 1/sqrt(32)
    logits[t] = s;
    atomicMaxF(&segmax[dst * HEADS + h], s);
}

// ---------------------------------------------------------------------------
// 3) exp(logit - max); accumulate un-normalized denom and exp*v[src] into agg.
//    (Normalization by denom deferred to finalize -> one fewer edge pass.)
// ---------------------------------------------------------------------------
__global__ void edge_scatter_kernel(const int* __restrict__ ei,
                                    const float* __restrict__ v,
                                    const float* __restrict__ logits,
                                    const float* __restrict__ segmax,
                                    float* __restrict__ denom,
                                    float* __restrict__ agg,
                                    int Ee)
{
    int t = blockIdx.x * blockDim.x + threadIdx.x;
    if (t >= Ee * HEADS) return;
    const int e = t >> 3;
    const int h = t & 7;
    const int src = ei[e];
    const int dst = ei[Ee + e];
    const float ex = expf(logits[t] - segmax[dst * HEADS + h]);
    atomicAdd(&denom[dst * HEADS + h], ex);
    const float4* vr = (const float4*)(v + (size_t)src * HID + h * DH);
    float* ar = agg + (size_t)dst * HID + h * DH;
    #pragma unroll
    for (int i = 0; i < DH / 4; ++i) {
        float4 vv = vr[i];
        atomicAdd(ar + 4 * i + 0, ex * vv.x);
        atomicAdd(ar + 4 * i + 1, ex * vv.y);
        atomicAdd(ar + 4 * i + 2, ex * vv.z);
        atomicAdd(ar + 4 * i + 3, ex * vv.w);
    }
}

// ---------------------------------------------------------------------------
// 4) finalize: agg/denom + skip -> exact GELU -> residual -> LayerNorm.
//    One 256-thread block per node; shared-memory tree reductions.
// ---------------------------------------------------------------------------
__global__ void finalize_kernel(const float* __restrict__ x,
                                const float* __restrict__ sk,
                                const float* __restrict__ denom,
                                const float* __restrict__ gamma,
                                const float* __restrict__ beta,
                                float* __restrict__ out)   // in: agg, out: result
{
    const int i = blockIdx.x;
    const int c = threadIdx.x;
    const size_t idx = (size_t)i * HID + c;

    const float dn = denom[i * HEADS + (c >> 5)];
    const float a  = (dn > 0.0f) ? out[idx] / dn : 0.0f;   // guard isolated nodes
    const float s  = a + sk[idx];
    const float g  = 0.5f * s * (1.0f + erff(s * 0.70710678118654752f)); // exact GELU
    const float y  = x[idx] + g;

    __shared__ float red[HID];
    __shared__ float mu_s, rstd_s;

    red[c] = y;
    __syncthreads();
    for (int off = HID / 2; off > 0; off >>= 1) {
        if (c < off) red[c] += red[c + off];
        __syncthreads();
    }
    if (c == 0) mu_s = red[0] * (1.0f / HID);
    __syncthreads();
    const float mu = mu_s;
    const float d  = y - mu;

    red[c] = d * d;
    __syncthreads();
    for (int off = HID / 2; off > 0; off >>= 1) {
        if (c < off) red[c] += red[c + off];
        __syncthreads();
    }
    if (c == 0) rstd_s = rsqrtf(red[0] * (1.0f / HID) + 1e-5f);
    __syncthreads();

    out[idx] = d * rstd_s * gamma[c] + beta[c];
}

// ---------------------------------------------------------------------------
extern "C" void kernel_launch(void* const* d_in, const int* in_sizes, int n_in,
                              void* d_out, int out_size, void* d_ws, size_t ws_size,
                              hipStream_t stream)
{
    const float* x     = (const float*)d_in[0];
    const int*   ei    = (const int*)d_in[1];
    const float* Wq    = (const float*)d_in[2];
    const float* bq    = (const float*)d_in[3];
    const float* Wk    = (const float*)d_in[4];
    const float* bk    = (const float*)d_in[5];
    const float* Wv    = (const float*)d_in[6];
    const float* bv    = (const float*)d_in[7];
    const float* Ws    = (const float*)d_in[8];
    const float* bs    = (const float*)d_in[9];
    const float* gamma = (const float*)d_in[10];
    const float* beta  = (const float*)d_in[11];

    const int Nn = in_sizes[0] / HID;   // 50000
    const int Ee = in_sizes[1] / 2;     // 800000

    // workspace layout (floats): q | k | v | skip | logits | segmax | denom
    float* ws     = (float*)d_ws;
    const size_t nrow = (size_t)Nn * HID;
    float* q      = ws;
    float* k      = q + nrow;
    float* v      = k + nrow;
    float* sk     = v + nrow;
    float* logits = sk + nrow;
    float* segmax = logits + (size_t)Ee * HEADS;
    float* denom  = segmax + (size_t)Nn * HEADS;
    float* agg    = (float*)d_out;      // accumulate aggregation directly in d_out

    {   // init
        const int total = (int)nrow;
        init_kernel<<<(total + 255) / 256, 256, 0, stream>>>(agg, segmax, denom,
                                                             total, Nn * HEADS);
    }
    {   // fused QKVS GEMM (WMMA + async global->LDS staging)
        dim3 grid((Nn + 15) / 16, 8);
        qkvs_gemm_kernel<<<grid, 256, 0, stream>>>(x, Wq, bq, Wk, bk, Wv, bv, Ws, bs,
                                                   q, k, v, sk, Nn);
    }
    {   // edge phase
        const int total = Ee * HEADS;
        const int blocks = (total + 255) / 256;
        edge_logits_kernel<<<blocks, 256, 0, stream>>>(ei, q, k, logits, segmax, Ee);
        edge_scatter_kernel<<<blocks, 256, 0, stream>>>(ei, v, logits, segmax, denom,
                                                        agg, Ee);
    }
    {   // epilogue
        finalize_kernel<<<Nn, 256, 0, stream>>>(x, sk, denom, gamma, beta, agg);
    }
}